// DeeperGCN_74457553043709
// MI455X (gfx1250) — compile-verified
//
#include <hip/hip_runtime.h>
#include <float.h>
#include <math.h>

typedef float v2f __attribute__((ext_vector_type(2)));
typedef float v8f __attribute__((ext_vector_type(8)));

#define H64 64
#define H128 128

// ---------------------------------------------------------------------------
// WMMA f32 GEMM:  Y[M,Nout] = (acc ? Y : 0) + X[M,K] @ W[K,Nout] + bias
// K / Nout are compile-time so the K-loop fully unrolls into a straight chain
// of v_wmma_f32_16x16x4_f32 with immediate-offset loads (no per-iter address
// math). One wave = one 16x16 output tile; 256 threads = 8 waves per block.
// A frag (16x4 f32, 2 VGPR): lane m = lane&15, k = 2*(lane>>4) + {0,1}
// B frag (4x16 f32, 2 VGPR): lane n = lane&15, k = 2*(lane>>4) + {0,1}
// C/D (16x16 f32, 8 VGPR):   row = v + 8*(lane>>4), col = lane&15
// ---------------------------------------------------------------------------
template <int K, int NOUT>
__global__ void wmma_gemm_kernel(const float* __restrict__ X,
                                 const float* __restrict__ Wm,
                                 const float* __restrict__ bias,
                                 float* __restrict__ Y,
                                 int M, int acc)
{
    const int lane = threadIdx.x & 31;
    const int waveInBlock = threadIdx.x >> 5;
    constexpr int nColTiles = NOUT >> 4;
    const long tile = (long)blockIdx.x * 8 + waveInBlock;
    const long totalTiles = (long)((M + 15) >> 4) * nColTiles;
    if (tile >= totalTiles) return;          // wave-uniform exit (EXEC stays full)

    const int rowTile = (int)(tile / nColTiles);
    const int colTile = (int)(tile % nColTiles);
    const int row0 = rowTile << 4;
    const int col0 = colTile << 4;

    const int mA   = lane & 15;
    const int kOff = (lane >> 4) << 1;       // 0 or 2
    const int nB   = lane & 15;

    int r = row0 + mA;
    if (r >= M) r = M - 1;                   // clamp (keeps EXEC uniform)
    const float* xrow = X + (size_t)r * K + kOff;
    const float* wcol = Wm + (size_t)kOff * NOUT + col0 + nB;

    v8f c;
#pragma unroll
    for (int i = 0; i < 8; ++i) c[i] = 0.0f;

#pragma unroll
    for (int k0 = 0; k0 < K; k0 += 4) {
        v2f a = *(const v2f*)(xrow + k0);    // contiguous 8B load, imm offset
        v2f b;
        b.x = wcol[(size_t)k0 * NOUT];
        b.y = wcol[(size_t)(k0 + 1) * NOUT];
        c = __builtin_amdgcn_wmma_f32_16x16x4_f32(
                /*neg_a=*/false, a, /*neg_b=*/false, b,
                /*c_mod=*/(short)0, c, /*reuse_a=*/false, /*reuse_b=*/false);
    }

    const float bb = bias[col0 + nB];
    const int mBase = (lane >> 4) << 3;
#pragma unroll
    for (int v = 0; v < 8; ++v) {
        const int rr = row0 + mBase + v;
        if (rr < M) {
            const size_t o = (size_t)rr * NOUT + col0 + nB;
            const float prev = acc ? Y[o] : 0.0f;
            Y[o] = prev + c[v] + bb;
        }
    }
}

// ---------------------------------------------------------------------------
// Row LayerNorm (+ optional ReLU). One wave per row; Wd/32 channels per lane,
// reduction via shfl_xor across the wave32.
// ---------------------------------------------------------------------------
__global__ void ln_relu_kernel(const float* __restrict__ in,
                               float* __restrict__ out,
                               const float* __restrict__ g,
                               const float* __restrict__ b,
                               int rows, int Wd, int doRelu)
{
    const int lane = threadIdx.x & 31;
    const int row  = (int)((blockIdx.x * blockDim.x + threadIdx.x) >> 5);
    if (row >= rows) return;

    const int vpl = Wd >> 5;                 // 2 (W=64) or 4 (W=128)
    const float* rp = in + (size_t)row * Wd;
    float vals[4];
    float s1 = 0.0f, s2 = 0.0f;
#pragma unroll 4
    for (int i = 0; i < 4; ++i) {
        if (i < vpl) {
            float v = rp[lane * vpl + i];
            vals[i] = v;
            s1 += v;
            s2 += v * v;
        }
    }
#pragma unroll
    for (int off = 16; off >= 1; off >>= 1) {
        s1 += __shfl_xor(s1, off, 32);
        s2 += __shfl_xor(s2, off, 32);
    }
    const float mu  = s1 / (float)Wd;
    const float var = s2 / (float)Wd - mu * mu;
    const float rstd = rsqrtf(var + 1e-5f);

    float* op = out + (size_t)row * Wd;
#pragma unroll 4
    for (int i = 0; i < 4; ++i) {
        if (i < vpl) {
            const int cch = lane * vpl + i;
            float y = (vals[i] - mu) * rstd * g[cch] + b[cch];
            if (doRelu) y = fmaxf(y, 0.0f);
            op[cch] = y;
        }
    }
}

// ---------------------------------------------------------------------------
// Per-layer scratch init: segmax = -FLT_MAX, denom = 0, agg = 0
// ---------------------------------------------------------------------------
__global__ void init_kernel(float* __restrict__ smax, float* __restrict__ den,
                            float* __restrict__ agg, size_t total)
{
    size_t i = (size_t)blockIdx.x * blockDim.x + threadIdx.x;
    if (i < total) {
        smax[i] = -FLT_MAX;
        den[i]  = 0.0f;
        agg[i]  = 0.0f;
    }
}

// float atomic max via sign-aware int/uint atomics (monotone toward max)
__device__ __forceinline__ void atomicMaxF(float* addr, float val)
{
    if (val >= 0.0f) atomicMax((int*)addr, __float_as_int(val));
    else             atomicMin((unsigned int*)addr, __float_as_uint(val));
}

// ---------------------------------------------------------------------------
// Edge pass 1: segment max of m = (relu(z[src]+e) + eps) * t over dst.
// One wave per edge, lane handles 2 channels (float2); e is streamed from HBM
// (with prefetch-ahead); z gathers and smax atomics stay L2-resident.
// ---------------------------------------------------------------------------
__global__ void edge_max_kernel(const float* __restrict__ z,
                                const float* __restrict__ e,
                                const int* __restrict__ ei,
                                const float* __restrict__ t_arr, int layer,
                                float* __restrict__ smax, int E)
{
    const int wave = (int)((blockIdx.x * blockDim.x + threadIdx.x) >> 5);
    const int lane = threadIdx.x & 31;
    if (wave >= E) return;

    if (wave + 64 < E)
        __builtin_prefetch(e + (size_t)(wave + 64) * H64, 0, 1);

    const int src = ei[wave];
    const int dst = ei[E + wave];
    const float tt = t_arr[layer];

    const float2 zv = *(const float2*)(z + (size_t)src * H64 + 2 * lane);
    const float2 ev = *(const float2*)(e + (size_t)wave * H64 + 2 * lane);

    const float m0 = (fmaxf(zv.x + ev.x, 0.0f) + 1e-7f) * tt;
    const float m1 = (fmaxf(zv.y + ev.y, 0.0f) + 1e-7f) * tt;

    float* sp = smax + (size_t)dst * H64 + 2 * lane;
    atomicMaxF(sp,     m0);
    atomicMaxF(sp + 1, m1);
}

// ---------------------------------------------------------------------------
// Edge pass 2 (fused): ex = exp(m - smax[dst]); denom += ex; agg += msg*ex.
// (alpha = ex/denom factors out of the aggregation -> divide per node later)
// ---------------------------------------------------------------------------
__global__ void edge_sum_kernel(const float* __restrict__ z,
                                const float* __restrict__ e,
                                const int* __restrict__ ei,
                                const float* __restrict__ t_arr, int layer,
                                const float* __restrict__ smax,
                                float* __restrict__ den,
                                float* __restrict__ agg, int E)
{
    const int wave = (int)((blockIdx.x * blockDim.x + threadIdx.x) >> 5);
    const int lane = threadIdx.x & 31;
    if (wave >= E) return;

    if (wave + 64 < E)
        __builtin_prefetch(e + (size_t)(wave + 64) * H64, 0, 1);

    const int src = ei[wave];
    const int dst = ei[E + wave];
    const float tt = t_arr[layer];

    const float2 zv = *(const float2*)(z + (size_t)src * H64 + 2 * lane);
    const float2 ev = *(const float2*)(e + (size_t)wave * H64 + 2 * lane);
    const float2 sm = *(const float2*)(smax + (size_t)dst * H64 + 2 * lane);

    const float msg0 = fmaxf(zv.x + ev.x, 0.0f) + 1e-7f;
    const float msg1 = fmaxf(zv.y + ev.y, 0.0f) + 1e-7f;
    const float ex0 = __expf(msg0 * tt - sm.x);
    const float ex1 = __expf(msg1 * tt - sm.y);

    const size_t o = (size_t)dst * H64 + 2 * lane;
    atomicAdd(den + o,     ex0);
    atomicAdd(den + o + 1, ex1);
    atomicAdd(agg + o,     msg0 * ex0);
    atomicAdd(agg + o + 1, msg1 * ex1);
}

// ---------------------------------------------------------------------------
// Node finalize (in place on agg): agg = z + agg / (denom + 1e-16)
// ---------------------------------------------------------------------------
__global__ void finalize_kernel(const float* __restrict__ z,
                                const float* __restrict__ den,
                                float* __restrict__ agg, size_t total)
{
    size_t i = (size_t)blockIdx.x * blockDim.x + threadIdx.x;
    if (i < total)
        agg[i] = z[i] + agg[i] / (den[i] + 1e-16f);
}

// ---------------------------------------------------------------------------
extern "C" void kernel_launch(void* const* d_in, const int* in_sizes, int n_in,
                              void* d_out, int out_size, void* d_ws, size_t ws_size,
                              hipStream_t stream)
{
    const float* x          = (const float*)d_in[0];
    const int*   edge_index = (const int*)  d_in[1];
    const float* edge_attr  = (const float*)d_in[2];
    const float* enc_node_w = (const float*)d_in[3];
    const float* enc_node_b = (const float*)d_in[4];
    const float* enc_edge_w = (const float*)d_in[5];
    const float* enc_edge_b = (const float*)d_in[6];
    const float* ln_g       = (const float*)d_in[7];
    const float* ln_b       = (const float*)d_in[8];
    const float* mlp1_w     = (const float*)d_in[9];
    const float* mlp1_b     = (const float*)d_in[10];
    const float* mlp_ln_g   = (const float*)d_in[11];
    const float* mlp_ln_b   = (const float*)d_in[12];
    const float* mlp2_w     = (const float*)d_in[13];
    const float* mlp2_b     = (const float*)d_in[14];
    const float* t_arr      = (const float*)d_in[15];

    const int N = in_sizes[0] / H64;      // 100000
    const int E = in_sizes[2] / H64;      // 1600000
    const int L = 4;

    float* ws   = (float*)d_ws;
    float* ebuf = ws;                                   // [E,64]
    float* hbuf = ebuf + (size_t)E * H64;               // [N,64]
    float* zbuf = hbuf + (size_t)N * H64;               // [N,64]
    float* smax = zbuf + (size_t)N * H64;               // [N,64]
    float* den  = smax + (size_t)N * H64;               // [N,64]
    float* agg  = den  + (size_t)N * H64;               // [N,64]
    float* t2h  = agg  + (size_t)N * H64;               // [N,128]

    auto blocksFor = [](int M, int nColTiles) {
        long tiles = (long)((M + 15) / 16) * nColTiles;
        return (int)((tiles + 7) / 8);
    };
    auto lnrelu = [&](const float* in, float* out, const float* g,
                      const float* b, int rows, int Wd, int relu) {
        int blocks = (rows + 7) / 8;                    // 8 waves per 256-thr block
        ln_relu_kernel<<<blocks, 256, 0, stream>>>(in, out, g, b, rows, Wd, relu);
    };

    const size_t nodeTot = (size_t)N * H64;
    const int nodeBlocks = (int)((nodeTot + 255) / 256);
    const int edgeBlocks = (E + 7) / 8;                 // 8 edges (waves) per block

    // Encoders (K=64 -> 64)
    wmma_gemm_kernel<64, 64><<<blocksFor(N, 4), 256, 0, stream>>>(
        x, enc_node_w, enc_node_b, hbuf, N, 0);
    wmma_gemm_kernel<64, 64><<<blocksFor(E, 4), 256, 0, stream>>>(
        edge_attr, enc_edge_w, enc_edge_b, ebuf, E, 0);

    for (int i = 0; i < L; ++i) {
        const float* zptr;
        if (i == 0) {
            zptr = hbuf;                                // bare conv, input = h
        } else {
            lnrelu(hbuf, zbuf, ln_g + i * H64, ln_b + i * H64, N, H64, 1);
            zptr = zbuf;                                // res+ pre-norm
        }

        init_kernel<<<nodeBlocks, 256, 0, stream>>>(smax, den, agg, nodeTot);
        edge_max_kernel<<<edgeBlocks, 256, 0, stream>>>(zptr, ebuf, edge_index,
                                                        t_arr, i, smax, E);
        edge_sum_kernel<<<edgeBlocks, 256, 0, stream>>>(zptr, ebuf, edge_index,
                                                        t_arr, i, smax, den, agg, E);
        finalize_kernel<<<nodeBlocks, 256, 0, stream>>>(zptr, den, agg, nodeTot);

        // MLP: Linear(64,128) -> LN -> ReLU -> Linear(128,64) (+ residual into h)
        wmma_gemm_kernel<64, 128><<<blocksFor(N, 8), 256, 0, stream>>>(
            agg, mlp1_w + (size_t)i * H64 * H128, mlp1_b + i * H128, t2h, N, 0);
        lnrelu(t2h, t2h, mlp_ln_g + i * H128, mlp_ln_b + i * H128, N, H128, 1);
        wmma_gemm_kernel<128, 64><<<blocksFor(N, 4), 256, 0, stream>>>(
            t2h, mlp2_w + (size_t)i * H128 * H64, mlp2_b + i * H64, hbuf, N,
            (i > 0) ? 1 : 0);
    }

    // Final: relu(LN(h, ln_g[0], ln_b[0]))
    lnrelu(hbuf, (float*)d_out, ln_g, ln_b, N, H64, 1);
}